// BCErrorCNN_15015205666866
// MI455X (gfx1250) — compile-verified
//
#include <hip/hip_runtime.h>
#include <math.h>

typedef float v2f __attribute__((ext_vector_type(2)));
typedef float v8f __attribute__((ext_vector_type(8)));

#define KSTAGE 96          // K rows staged in LDS per step (48KB)
#define NSTAGES 4          // 384 / KSTAGE

// Wave-level 16x384 @ 384x128 f32 GEMM using V_WMMA_F32_16X16X4_F32.
// Block = 512 threads (16 waves); each wave computes a 16(row)x128(col) tile.
// Weights staged through LDS in 4 chunks of 96x128, K-pair interleaved so a
// B fragment is one aligned ds_load_b64; odd pairs xor-swizzled by 32 words
// to keep the two half-waves on disjoint LDS banks.
__device__ __forceinline__ void gemm_tile(
    const float* __restrict__ Arow,   // per-lane pointer: &A[m][0] for this lane's row m
    const float* __restrict__ W,      // 384x128 weights in global memory (row-major K x N)
    float* sW,                        // KSTAGE*128 floats of LDS, pair-interleaved
    v8f acc[8], int koff, int nl, int tid)
{
  for (int s = 0; s < NSTAGES; ++s) {
    __syncthreads();  // all waves done reading previous stage
    for (int i = tid; i < (KSTAGE / 2) * 128; i += 512) {
      const int kp = i >> 7;          // K-pair index 0..47
      const int n  = i & 127;         // column
      const float* g = W + (size_t)(s * KSTAGE + kp * 2) * 128 + n;
      v2f w; w.x = g[0]; w.y = g[128];                       // rows 2kp, 2kp+1
      *(v2f*)(sW + kp * 256 + ((n * 2) ^ ((kp & 1) << 5))) = w;
    }
    __syncthreads();
    const float* As = Arow + s * KSTAGE;
    for (int kt = 0; kt < KSTAGE / 4; ++kt) {
      const int kl = kt * 4 + koff;   // lanes<16: K=4kt(,+1); lanes>=16: K=4kt+2(,+3)
      v2f a = *(const v2f*)(As + kl);
      const int kp = kl >> 1;
      const float* brow = sW + kp * 256;
      const int sw = (kp & 1) << 5;
      #pragma unroll
      for (int nt = 0; nt < 8; ++nt) {
        v2f b = *(const v2f*)(brow + ((((nt * 16 + nl) << 1)) ^ sw));
        acc[nt] = __builtin_amdgcn_wmma_f32_16x16x4_f32(
            false, a, false, b, (short)0, acc[nt], false, false);
      }
    }
  }
}

// ---------------- Stage 1: conv (K=3,C=9 -> F=128) + relu ----------------
__global__ __launch_bounds__(256)
void conv_kernel(const float* __restrict__ in, const float* __restrict__ cw,
                 const float* __restrict__ cb, float* __restrict__ x1, int c0)
{
  __shared__ float sw[27 * 128];
  __shared__ float sb[128];
  const int tid = threadIdx.x;
  for (int i = tid; i < 27 * 128; i += 256) sw[i] = cw[i];
  if (tid < 128) sb[tid] = cb[tid];
  __syncthreads();
  const int gid = blockIdx.x * 256 + tid;      // over CN*15*128
  const int f = gid & 127;
  const int t = (gid >> 7) % 15;
  const int bl = gid / (15 * 128);
  const float* ip = in + (size_t)(c0 + bl) * 153 + t * 9;  // 27 contiguous floats
  float s = sb[f];
  #pragma unroll
  for (int u = 0; u < 27; ++u) s = fmaf(ip[u], sw[u * 128 + f], s);
  x1[gid] = s > 0.f ? s : 0.f;
}

// ------- Stage 2: lc1 (12 positions) + relu, fused pairwise max-pool -------
// grid = (CN/256, 6); block = 512. Output x3[row, j, 128].
__global__ __launch_bounds__(512)
void lc1_pool_kernel(const float* __restrict__ x1, const float* __restrict__ lc1w,
                     const float* __restrict__ lc1b, float* __restrict__ x3)
{
  __shared__ float sW[KSTAGE * 128];
  const int tid = threadIdx.x;
  const int wave = tid >> 5, lane = tid & 31;
  const int nl = lane & 15;
  const int koff = (lane >> 4) << 1;
  const int j = blockIdx.y;                    // pooled position 0..5
  const int row0 = blockIdx.x * 256 + wave * 16;
  const int m = row0 + nl;
  v8f keep[8], acc[8];
  for (int pass = 0; pass < 2; ++pass) {
    const int l = 2 * j + pass;
    const float* Arow = x1 + (size_t)m * 1920 + l * 128;  // patch l is contiguous
    #pragma unroll
    for (int nt = 0; nt < 8; ++nt) { v8f z = {}; acc[nt] = z; }
    gemm_tile(Arow, lc1w + (size_t)l * 49152, sW, acc, koff, nl, tid);
    #pragma unroll
    for (int nt = 0; nt < 8; ++nt) {
      const float bv = lc1b[l * 128 + nt * 16 + nl];
      #pragma unroll
      for (int r = 0; r < 8; ++r) {
        float v = acc[nt][r] + bv;
        acc[nt][r] = v > 0.f ? v : 0.f;
      }
    }
    if (pass == 0) {
      #pragma unroll
      for (int nt = 0; nt < 8; ++nt) keep[nt] = acc[nt];
    }
  }
  const int rbase = row0 + ((lane >> 4) << 3);  // D layout: VGPR r -> row r (+8 for hi lanes)
  #pragma unroll
  for (int nt = 0; nt < 8; ++nt) {
    #pragma unroll
    for (int r = 0; r < 8; ++r) {
      float a = keep[nt][r], b = acc[nt][r];
      x3[((size_t)(rbase + r) * 6 + j) * 128 + nt * 16 + nl] = a > b ? a : b;
    }
  }
}

// --------- Stage 3: lc2 (4 positions) + relu, fused mean over l ---------
__global__ __launch_bounds__(512)
void lc2_mean_kernel(const float* __restrict__ x3, const float* __restrict__ lc2w,
                     const float* __restrict__ lc2b, float* __restrict__ xm)
{
  __shared__ float sW[KSTAGE * 128];
  const int tid = threadIdx.x;
  const int wave = tid >> 5, lane = tid & 31;
  const int nl = lane & 15;
  const int koff = (lane >> 4) << 1;
  const int row0 = blockIdx.x * 256 + wave * 16;
  const int m = row0 + nl;
  v8f sum[8], acc[8];
  #pragma unroll
  for (int nt = 0; nt < 8; ++nt) { v8f z = {}; sum[nt] = z; }
  for (int l = 0; l < 4; ++l) {
    const float* Arow = x3 + (size_t)m * 768 + l * 128;
    #pragma unroll
    for (int nt = 0; nt < 8; ++nt) { v8f z = {}; acc[nt] = z; }
    gemm_tile(Arow, lc2w + (size_t)l * 49152, sW, acc, koff, nl, tid);
    #pragma unroll
    for (int nt = 0; nt < 8; ++nt) {
      const float bv = lc2b[l * 128 + nt * 16 + nl];
      #pragma unroll
      for (int r = 0; r < 8; ++r) {
        float v = acc[nt][r] + bv;
        sum[nt][r] += (v > 0.f ? v : 0.f);
      }
    }
  }
  const int rbase = row0 + ((lane >> 4) << 3);
  #pragma unroll
  for (int nt = 0; nt < 8; ++nt)
    #pragma unroll
    for (int r = 0; r < 8; ++r)
      xm[(size_t)(rbase + r) * 128 + nt * 16 + nl] = sum[nt][r] * 0.25f;
}

// ------------- Stage 4: dense 128->100 relu, 100->1 sigmoid -------------
__global__ __launch_bounds__(256)
void head_kernel(const float* __restrict__ xm, const float* __restrict__ d1w,
                 const float* __restrict__ d1b, const float* __restrict__ d2w,
                 float* __restrict__ out, int c0)
{
  __shared__ float s1[128 * 100];
  __shared__ float sb1[128];
  __shared__ float s2[128];
  const int tid = threadIdx.x;
  for (int i = tid; i < 128 * 100; i += 256) s1[i] = d1w[i];
  if (tid < 100) { sb1[tid] = d1b[tid]; s2[tid] = d2w[tid]; }
  __syncthreads();
  const int r = blockIdx.x * 256 + tid;
  const float* xr = xm + (size_t)r * 128;
  float h[100];
  #pragma unroll
  for (int jj = 0; jj < 100; ++jj) h[jj] = sb1[jj];
  for (int k = 0; k < 128; ++k) {
    const float xv = xr[k];
    #pragma unroll
    for (int jj = 0; jj < 100; ++jj) h[jj] = fmaf(xv, s1[k * 100 + jj], h[jj]);
  }
  float z = 0.f;
  #pragma unroll
  for (int jj = 0; jj < 100; ++jj) z = fmaf(h[jj] > 0.f ? h[jj] : 0.f, s2[jj], z);
  out[c0 + r] = 1.0f / (1.0f + expf(-z));
}

extern "C" void kernel_launch(void* const* d_in, const int* in_sizes, int n_in,
                              void* d_out, int out_size, void* d_ws, size_t ws_size,
                              hipStream_t stream)
{
  (void)n_in; (void)out_size;
  const float* in  = (const float*)d_in[0];
  const float* cw  = (const float*)d_in[1];
  const float* cb  = (const float*)d_in[2];
  const float* l1w = (const float*)d_in[3];
  const float* l1b = (const float*)d_in[4];
  const float* l2w = (const float*)d_in[5];
  const float* l2b = (const float*)d_in[6];
  const float* d1w = (const float*)d_in[7];
  const float* d1b = (const float*)d_in[8];
  const float* d2w = (const float*)d_in[9];
  float* out = (float*)d_out;

  const int B = in_sizes[0] / (17 * 9);   // 32768
  // Largest power-of-two chunk whose scratch (x1 + x3 + xm = 2816 floats/row) fits.
  int PC = 32768;
  if (PC > B) PC = B;
  while (PC > 256 && (size_t)PC * 2816ull * 4ull > ws_size) PC >>= 1;

  float* x1 = (float*)d_ws;                       // [PC, 15, 128]
  float* x3 = x1 + (size_t)PC * 1920;             // [PC, 6, 128]
  float* xm = x3 + (size_t)PC * 768;              // [PC, 128]

  for (int c0 = 0; c0 < B; c0 += PC) {
    int CN = (B - c0 < PC) ? (B - c0) : PC;
    conv_kernel<<<(CN * 1920) / 256, 256, 0, stream>>>(in, cw, cb, x1, c0);
    lc1_pool_kernel<<<dim3(CN / 256, 6), 512, 0, stream>>>(x1, l1w, l1b, x3);
    lc2_mean_kernel<<<CN / 256, 512, 0, stream>>>(x3, l2w, l2b, xm);
    head_kernel<<<CN / 256, 256, 0, stream>>>(xm, d1w, d1b, d2w, out, c0);
  }
}